// SimCAM_87359634801247
// MI455X (gfx1250) — compile-verified
//
#include <hip/hip_runtime.h>

// Problem constants (match reference).
#define HH   64
#define WW   64
#define CCH  2048          // channels (K dimension)
#define HW   4096          // H*W     (M and N dimension)
#define IMG  224

typedef __attribute__((ext_vector_type(2))) float v2f;
typedef __attribute__((ext_vector_type(8))) float v8f;

#define WMMA_F32(A, B, C) \
    __builtin_amdgcn_wmma_f32_16x16x4_f32(false, (A), false, (B), (short)0, (C), false, false)

// ---- monotonic float<->uint mapping so we can use integer atomicMax ----
__device__ __forceinline__ unsigned f2ord(float f) {
    unsigned u = __float_as_uint(f);
    return (u & 0x80000000u) ? ~u : (u | 0x80000000u);
}
__device__ __forceinline__ float ord2f(unsigned s) {
    unsigned b = (s & 0x80000000u) ? (s ^ 0x80000000u) : ~s;
    return __uint_as_float(b);
}

// ------------------------------------------------------------------
// Kernel 1: per-channel column sums of x0 and x1 (S0[c], S1[c]),
// float4-vectorized (b128 loads). Also initializes the global-max
// slot (runs before gemm_max in stream order -> replay-safe).
// 4 blocks x 256 threads = 1024 threads, 4 channels each.
// ------------------------------------------------------------------
__global__ __launch_bounds__(256)
void simcam_colsum(const float* __restrict__ x0, const float* __restrict__ x1,
                   float* __restrict__ S0, float* __restrict__ S1,
                   unsigned* __restrict__ maxp) {
    const int t = blockIdx.x * 256 + threadIdx.x;      // 0..1023
    if (t == 0) maxp[0] = 0u;                          // ordered "-inf"
    const bool second = t >= 512;
    const float* __restrict__ X = second ? x1 : x0;
    float* __restrict__ S = second ? S1 : S0;
    const int c4 = (t & 511) * 4;
    float sx = 0.f, sy = 0.f, sz = 0.f, sw = 0.f;
    #pragma unroll 4
    for (int r = 0; r < HW; ++r) {
        float4 v = *reinterpret_cast<const float4*>(X + (size_t)r * CCH + c4);
        sx += v.x; sy += v.y; sz += v.z; sw += v.w;
    }
    float4 o; o.x = sx; o.y = sy; o.z = sz; o.w = sw;
    *reinterpret_cast<float4*>(S + c4) = o;
}

// ------------------------------------------------------------------
// Kernel 2: max over D = X0 * X1^T using V_WMMA_F32_16X16X4_F32.
// Block = 256 threads = 8 waves, block tile 128x128.
// Wave (wm = wid&3, wn = wid>>2) owns a 32x64 strip: 2 A-fragments
// x 4 B-fragments -> 8 fp32 accumulators. Fragment loads for step
// c+4 are issued before the WMMAs consuming step c (software
// pipeline) so loads stay in flight under the matrix ops.
// Fragment layout (ISA 7.12.2, 32-bit A 16x4): lane L<16 holds
// row M=L with K={0,1}; lane 16+L holds row M=L with K={2,3};
// same striping for B taken from row-major x1 (B = X1^T).
// ------------------------------------------------------------------
__global__ __launch_bounds__(256)
void simcam_gemm_max(const float* __restrict__ x0, const float* __restrict__ x1,
                     unsigned* __restrict__ maxp) {
    const int wid  = threadIdx.x >> 5;
    const int lane = threadIdx.x & 31;
    const int r    = lane & 15;
    const int h    = lane >> 4;          // 0: K={0,1}, 1: K={2,3}

    const int m0 = blockIdx.x * 128 + (wid & 3) * 32;
    const int n0 = blockIdx.y * 128 + (wid >> 2) * 64;

    const float* __restrict__ A0 = x0 + (size_t)(m0 + r) * CCH + 2 * h;
    const float* __restrict__ A1 = A0 + (size_t)16 * CCH;
    const float* __restrict__ B0 = x1 + (size_t)(n0 + r) * CCH + 2 * h;
    const float* __restrict__ B1 = B0 + (size_t)16 * CCH;
    const float* __restrict__ B2 = B0 + (size_t)32 * CCH;
    const float* __restrict__ B3 = B0 + (size_t)48 * CCH;

    v8f acc00 = {}, acc01 = {}, acc02 = {}, acc03 = {};
    v8f acc10 = {}, acc11 = {}, acc12 = {}, acc13 = {};

    // Prologue: fragments for k-step 0.
    v2f a0 = *reinterpret_cast<const v2f*>(A0);
    v2f a1 = *reinterpret_cast<const v2f*>(A1);
    v2f b0 = *reinterpret_cast<const v2f*>(B0);
    v2f b1 = *reinterpret_cast<const v2f*>(B1);
    v2f b2 = *reinterpret_cast<const v2f*>(B2);
    v2f b3 = *reinterpret_cast<const v2f*>(B3);

    #pragma unroll 2
    for (int c = 4; c < CCH; c += 4) {
        // Issue next k-step's loads first (stay in flight under WMMAs).
        v2f na0 = *reinterpret_cast<const v2f*>(A0 + c);
        v2f na1 = *reinterpret_cast<const v2f*>(A1 + c);
        v2f nb0 = *reinterpret_cast<const v2f*>(B0 + c);
        v2f nb1 = *reinterpret_cast<const v2f*>(B1 + c);
        v2f nb2 = *reinterpret_cast<const v2f*>(B2 + c);
        v2f nb3 = *reinterpret_cast<const v2f*>(B3 + c);

        acc00 = WMMA_F32(a0, b0, acc00);
        acc01 = WMMA_F32(a0, b1, acc01);
        acc02 = WMMA_F32(a0, b2, acc02);
        acc03 = WMMA_F32(a0, b3, acc03);
        acc10 = WMMA_F32(a1, b0, acc10);
        acc11 = WMMA_F32(a1, b1, acc11);
        acc12 = WMMA_F32(a1, b2, acc12);
        acc13 = WMMA_F32(a1, b3, acc13);

        a0 = na0; a1 = na1;
        b0 = nb0; b1 = nb1; b2 = nb2; b3 = nb3;
    }

    // Epilogue: last k-step.
    acc00 = WMMA_F32(a0, b0, acc00);
    acc01 = WMMA_F32(a0, b1, acc01);
    acc02 = WMMA_F32(a0, b2, acc02);
    acc03 = WMMA_F32(a0, b3, acc03);
    acc10 = WMMA_F32(a1, b0, acc10);
    acc11 = WMMA_F32(a1, b1, acc11);
    acc12 = WMMA_F32(a1, b2, acc12);
    acc13 = WMMA_F32(a1, b3, acc13);

    // Lane-local max over all 64 accumulated elements.
    float m = -3.402823466e38f;
    #pragma unroll
    for (int e = 0; e < 8; ++e) {
        m = fmaxf(m, acc00[e]); m = fmaxf(m, acc01[e]);
        m = fmaxf(m, acc02[e]); m = fmaxf(m, acc03[e]);
        m = fmaxf(m, acc10[e]); m = fmaxf(m, acc11[e]);
        m = fmaxf(m, acc12[e]); m = fmaxf(m, acc13[e]);
    }
    // Wave32 butterfly reduction.
    #pragma unroll
    for (int off = 16; off >= 1; off >>= 1)
        m = fmaxf(m, __shfl_xor(m, off, 32));
    if (lane == 0) atomicMax(maxp, f2ord(m));
}

// ------------------------------------------------------------------
// Kernel 3: saliency maps.
//   d1[i] = dot(x0[i,:], S1) / max      (sum over target positions)
//   d2[i] = dot(x1[i,:], S0) / max      (sum over query positions)
// 32 blocks x 256 threads; blocks 0..15 -> d1, 16..31 -> d2.
// S vector staged in LDS (8 KB).
// ------------------------------------------------------------------
__global__ __launch_bounds__(256)
void simcam_saliency(const float* __restrict__ x0, const float* __restrict__ x1,
                     const float* __restrict__ S0, const float* __restrict__ S1,
                     const unsigned* __restrict__ maxp,
                     float* __restrict__ d1, float* __restrict__ d2) {
    __shared__ float s[CCH];
    const bool second = blockIdx.x >= 16;
    const float* __restrict__ S = second ? S0 : S1;
    const float* __restrict__ X = second ? x1 : x0;
    float* __restrict__ D       = second ? d2 : d1;
    const int row = ((int)blockIdx.x & 15) * 256 + threadIdx.x;   // 0..4095

    for (int c = threadIdx.x; c < CCH; c += 256) s[c] = S[c];
    __syncthreads();

    const float* __restrict__ xr = X + (size_t)row * CCH;
    float acc = 0.0f;
    #pragma unroll 4
    for (int c = 0; c < CCH; c += 4) {
        float4 xv = *reinterpret_cast<const float4*>(xr + c);     // b128 load
        acc += xv.x * s[c] + xv.y * s[c + 1] + xv.z * s[c + 2] + xv.w * s[c + 3];
    }
    const float inv = 1.0f / ord2f(maxp[0]);
    D[row] = acc * inv;
}

// ------------------------------------------------------------------
// Kernel 4: half-pixel bilinear resize [2,64,64] -> [2,224,224]
// (matches jax.image.resize 'bilinear' upsampling semantics).
// ------------------------------------------------------------------
__global__ __launch_bounds__(256)
void simcam_resize(const float* __restrict__ d1, const float* __restrict__ d2,
                   float* __restrict__ out) {
    const int idx = blockIdx.x * 256 + threadIdx.x;
    if (idx >= 2 * IMG * IMG) return;
    const int s   = idx / (IMG * IMG);
    const int rem = idx - s * IMG * IMG;
    const int y   = rem / IMG;
    const int x   = rem - y * IMG;
    const float* __restrict__ map = s ? d2 : d1;

    const float scale = (float)HH / (float)IMG;
    float fy = ((float)y + 0.5f) * scale - 0.5f;
    float fx = ((float)x + 0.5f) * scale - 0.5f;
    fy = fminf(fmaxf(fy, 0.0f), (float)(HH - 1));
    fx = fminf(fmaxf(fx, 0.0f), (float)(WW - 1));
    const int y0 = (int)fy, x0 = (int)fx;
    const int y1 = min(y0 + 1, HH - 1), x1 = min(x0 + 1, WW - 1);
    const float wy = fy - (float)y0, wx = fx - (float)x0;

    const float v00 = map[y0 * WW + x0], v01 = map[y0 * WW + x1];
    const float v10 = map[y1 * WW + x0], v11 = map[y1 * WW + x1];
    out[idx] = (1.0f - wy) * ((1.0f - wx) * v00 + wx * v01)
             +          wy * ((1.0f - wx) * v10 + wx * v11);
}

// ------------------------------------------------------------------
extern "C" void kernel_launch(void* const* d_in, const int* in_sizes, int n_in,
                              void* d_out, int out_size, void* d_ws, size_t ws_size,
                              hipStream_t stream) {
    (void)in_sizes; (void)n_in; (void)out_size; (void)ws_size;

    const float* feats = (const float*)d_in[0];          // [2, 64, 64, 2048] fp32
    const float* x0 = feats;                             // query  [4096, 2048]
    const float* x1 = feats + (size_t)HW * CCH;          // target [4096, 2048]
    float* out = (float*)d_out;                          // [2, 224, 224] fp32

    // Workspace layout (floats): [0]=ordered max (as uint), then S0, S1, d1, d2.
    float*    wsf  = (float*)d_ws;
    unsigned* maxp = (unsigned*)wsf;
    float* S0  = wsf + 64;
    float* S1  = S0 + CCH;
    float* d1m = S1 + CCH;
    float* d2m = d1m + HW;

    // 1) Column sums + max-slot init.
    simcam_colsum<<<dim3(4), dim3(256), 0, stream>>>(x0, x1, S0, S1, maxp);

    // 2) WMMA fp32 GEMM-max over the 4096x4096 correlation volume.
    //    Grid: 32x32 block tiles of 128x128.
    simcam_gemm_max<<<dim3(32, 32), dim3(256), 0, stream>>>(x0, x1, maxp);

    // 3) Saliency maps d1, d2 (normalized by the global max).
    simcam_saliency<<<dim3(32), dim3(256), 0, stream>>>(x0, x1, S0, S1, maxp, d1m, d2m);

    // 4) Bilinear upsample to [2, 224, 224].
    const int nout = 2 * IMG * IMG;
    simcam_resize<<<dim3((nout + 255) / 256), dim3(256), 0, stream>>>(d1m, d2m, out);
}